// GCN_55473797595452
// MI455X (gfx1250) — compile-verified
//
#include <hip/hip_runtime.h>

typedef float v2f __attribute__((ext_vector_type(2)));
typedef float v8f __attribute__((ext_vector_type(8)));

#define BATCH 2
#define NNODE 50000
#define FEAT  128          // D == H == 128
#define MROWS (BATCH * NNODE)   // 100000, divisible by 16; batch boundary tile-aligned

// ---------------------------------------------------------------------------
// init kernels
// ---------------------------------------------------------------------------
__global__ void zero_f32_kernel(float* __restrict__ p, int n) {
  int i = blockIdx.x * blockDim.x + threadIdx.x;
  if (i < n) p[i] = 0.0f;
}

__global__ void init_out_kernel(float* __restrict__ out, const float* __restrict__ b3) {
  int i = threadIdx.x;
  if (i < BATCH * 10) out[i] = b3[i % 10];   // mean(x + b3) = mean(x) + b3
}

// ---------------------------------------------------------------------------
// Generic fp32 WMMA GEMM:  C[M x N] = act(A[M x K] @ B[K x N] + bias)
// Row-major. M%16==0, N%16==0, K%4==0. One wave (32 threads) per 16x16 tile.
// act: 0 = identity, 1 = relu
// ---------------------------------------------------------------------------
__global__ __launch_bounds__(32)
void wmma_gemm_f32_kernel(const float* __restrict__ A, const float* __restrict__ Bm,
                          const float* __restrict__ bias, float* __restrict__ C,
                          int K, int N, int act) {
  const int lane = threadIdx.x;        // 0..31 (wave32)
  const int half = lane >> 4;          // 0 or 1
  const int mn   = lane & 15;
  const int mBase = blockIdx.x * 16;
  const int nBase = blockIdx.y * 16;

  const float* arow = A  + (size_t)(mBase + mn) * K;   // lanes 0-15 & 16-31: same M rows
  const float* bcol = Bm + nBase + mn;

  v8f c = {};
  for (int kb = 0; kb < K; kb += 4) {
    v2f a, b;
    // A 16x4 frag: VGPR0 = K(2*half), VGPR1 = K(2*half+1)
    a[0] = arow[kb + 2 * half + 0];
    a[1] = arow[kb + 2 * half + 1];
    // B 4x16 frag: VGPR v = K(2*half + v), N = lane%16
    b[0] = bcol[(size_t)(kb + 2 * half + 0) * N];
    b[1] = bcol[(size_t)(kb + 2 * half + 1) * N];
    c = __builtin_amdgcn_wmma_f32_16x16x4_f32(false, a, false, b, (short)0, c,
                                              false, false);
  }

  const int col = nBase + mn;
  const float bv = bias ? bias[col] : 0.0f;
#pragma unroll
  for (int i = 0; i < 8; ++i) {
    int row = mBase + i + 8 * half;    // C/D layout: VGPR i -> M = i (+8 for upper half)
    float v = c[i] + bv;
    if (act == 1) v = fmaxf(v, 0.0f);
    C[(size_t)row * N + col] = v;
  }
}

// ---------------------------------------------------------------------------
// Edge scatter-add: agg[b, dst, f] += support[b, src, f]  (both batches/thread)
// One thread per (edge, feature). Native f32 atomics; agg+support live in L2.
// ---------------------------------------------------------------------------
__global__ void scatter_add_kernel(const float* __restrict__ support,
                                   const int* __restrict__ src,
                                   const int* __restrict__ dst,
                                   float* __restrict__ agg, int E) {
  int tid = blockIdx.x * blockDim.x + threadIdx.x;
  int e = tid >> 7;          // 128 features
  int f = tid & 127;
  if (e < E) {
    int s = src[e], d = dst[e];
    float v0 = support[(size_t)s * FEAT + f];
    float v1 = support[((size_t)NNODE + s) * FEAT + f];
    __hip_atomic_fetch_add(&agg[(size_t)d * FEAT + f], v0,
                           __ATOMIC_RELAXED, __HIP_MEMORY_SCOPE_AGENT);
    __hip_atomic_fetch_add(&agg[((size_t)NNODE + d) * FEAT + f], v1,
                           __ATOMIC_RELAXED, __HIP_MEMORY_SCOPE_AGENT);
  }
}

// ---------------------------------------------------------------------------
// h = gelu(agg + b), exact gelu (erf), in place
// ---------------------------------------------------------------------------
__global__ void bias_gelu_kernel(float* __restrict__ h, const float* __restrict__ bias,
                                 int total) {
  int i = blockIdx.x * blockDim.x + threadIdx.x;
  if (i < total) {
    float x = h[i] + bias[i & (FEAT - 1)];
    h[i] = 0.5f * x * (1.0f + erff(x * 0.70710678118654752440f));
  }
}

// ---------------------------------------------------------------------------
// Final layer: out[b, :10] += mean_n( h2[b,n,:32] @ W3[32x10] )
// One wave per 16-row tile (tile never crosses the batch boundary:
// 50000 % 16 == 0). N padded 10 -> 16 with zero columns; in-wave row
// reduction, then 10 atomics per tile.
// ---------------------------------------------------------------------------
__global__ __launch_bounds__(32)
void wmma_final_kernel(const float* __restrict__ A /* M x 32 */,
                       const float* __restrict__ W3 /* 32 x 10 */,
                       float* __restrict__ out /* 2 x 10 */) {
  const int lane = threadIdx.x;
  const int half = lane >> 4;
  const int mn   = lane & 15;
  const int mBase = blockIdx.x * 16;
  const int K = 32, NO = 10;

  const float* arow = A + (size_t)(mBase + mn) * K;

  v8f c = {};
  for (int kb = 0; kb < K; kb += 4) {
    v2f a, b;
    a[0] = arow[kb + 2 * half + 0];
    a[1] = arow[kb + 2 * half + 1];
    b[0] = (mn < NO) ? W3[(size_t)(kb + 2 * half + 0) * NO + mn] : 0.0f;
    b[1] = (mn < NO) ? W3[(size_t)(kb + 2 * half + 1) * NO + mn] : 0.0f;
    c = __builtin_amdgcn_wmma_f32_16x16x4_f32(false, a, false, b, (short)0, c,
                                              false, false);
  }

  // reduce the 8 rows held by this lane, then combine the two half-waves
  float s = 0.0f;
#pragma unroll
  for (int i = 0; i < 8; ++i) s += c[i];
  s += __shfl_xor(s, 16, 32);          // rows mBase..mBase+15 summed, col = mn

  if (half == 0 && mn < NO) {
    const int batch = (mBase >= NNODE) ? 1 : 0;
    __hip_atomic_fetch_add(&out[batch * NO + mn], s * (1.0f / (float)NNODE),
                           __ATOMIC_RELAXED, __HIP_MEMORY_SCOPE_AGENT);
  }
}

// ---------------------------------------------------------------------------
// launch
// ---------------------------------------------------------------------------
extern "C" void kernel_launch(void* const* d_in, const int* in_sizes, int n_in,
                              void* d_out, int out_size, void* d_ws, size_t ws_size,
                              hipStream_t stream) {
  const float* x    = (const float*)d_in[0];
  const int*   edge = (const int*)d_in[1];
  const float* W    = (const float*)d_in[2];
  const float* b    = (const float*)d_in[3];
  const float* W1   = (const float*)d_in[4];
  const float* b1   = (const float*)d_in[5];
  const float* W2   = (const float*)d_in[6];
  const float* b2   = (const float*)d_in[7];
  const float* W3   = (const float*)d_in[8];
  const float* b3   = (const float*)d_in[9];
  float* out = (float*)d_out;

  const int E = in_sizes[1] / 2;         // edge_index is (2, E)
  const int M = MROWS;                   // 100000 rows (batch-major)

  // workspace layout (peak 102.4 MB):
  //   region0 [0,              M*128) : support, later h1 (M*64)
  //   region1 [M*128,        2*M*128) : agg/h,  later h2 (M*32)
  float* region0 = (float*)d_ws;
  float* region1 = region0 + (size_t)M * FEAT;
  float* support = region0;
  float* agg     = region1;
  float* h1      = region0;
  float* h2      = region1;

  const int totalBNH = M * FEAT;         // 12.8M

  // 1) init: zero agg, seed out with b3
  zero_f32_kernel<<<(totalBNH + 255) / 256, 256, 0, stream>>>(agg, totalBNH);
  init_out_kernel<<<1, 32, 0, stream>>>(out, b3);

  // 2) support = x @ W           (M x 128) = (M x 128)(128 x 128)
  wmma_gemm_f32_kernel<<<dim3(M / 16, FEAT / 16), 32, 0, stream>>>(
      x, W, nullptr, support, FEAT, FEAT, 0);

  // 3) agg[:, dst, :] += support[:, src, :]
  {
    long long threads = (long long)E * FEAT;
    scatter_add_kernel<<<(unsigned)((threads + 255) / 256), 256, 0, stream>>>(
        support, edge /*src*/, edge + E /*dst*/, agg, E);
  }

  // 4) h = gelu(agg + b)  (in place, exact gelu)
  bias_gelu_kernel<<<(totalBNH + 255) / 256, 256, 0, stream>>>(agg, b, totalBNH);

  // 5) h1 = relu(h @ W1 + b1)    (M x 64)
  wmma_gemm_f32_kernel<<<dim3(M / 16, 64 / 16), 32, 0, stream>>>(
      agg, W1, b1, h1, FEAT, 64, 1);

  // 6) h2 = relu(h1 @ W2 + b2)   (M x 32)
  wmma_gemm_f32_kernel<<<dim3(M / 16, 32 / 16), 32, 0, stream>>>(
      h1, W2, b2, h2, 64, 32, 1);

  // 7) out += mean_n(h2 @ W3)    (2 x 10)
  wmma_final_kernel<<<M / 16, 32, 0, stream>>>(h2, W3, out);
}